// Cross_head_65549790871667
// MI455X (gfx1250) — compile-verified
//
#include <hip/hip_runtime.h>

// CDNA5 / gfx1250: fp32 GEMM per head via V_WMMA_F32_16X16X4_F32, with
// Tensor Data Mover (tensor_load_to_lds) double-buffered LDS staging and
// software-pipelined fragment loads (ds latency hidden behind 12 WMMAs).
//
// att[c,l,k] = (sum_m attn[c,l,m] * patchesT[c,m,k]) / nz[c,l]
// out = leaky_relu((att+1)*x, 0.2) written in fold (col2im) order.

typedef __attribute__((ext_vector_type(2))) float    v2f;
typedef __attribute__((ext_vector_type(8))) float    v8f;
typedef __attribute__((ext_vector_type(4))) unsigned v4u;
typedef __attribute__((ext_vector_type(4))) int      v4i;
typedef __attribute__((ext_vector_type(8))) int      v8i;

#define C_HEADS 4
#define L_DIM   4608
#define NPIX    81
#define HP_DIM  72
#define WP_DIM  5184
#define PSZ     9
#define NWB     576
#define BK      32
#define NTILES  144        // 4608 / 32 K-tiles
#define ROWS_WG 64         // 2 waves x 32 rows

// A tile in LDS: 64 rows x 32 floats, row stride 36 floats (TDM pad 4/32)
#define A_STR     36
#define A_TILE_B  (64 * A_STR * 4)           // 9216 B
// B tile: 16 row-pairs x 112 float2 (96 used + 16 pad), fragment-interleaved
#define B_RP_STR  112                        // float2 stride per row-pair
#define B_TILE_F  (16 * B_RP_STR * 2)        // 3584 floats
#define B_TILE_B  (B_TILE_F * 4)             // 14336 B
#define B_BASE    (2 * A_TILE_B)             // after two A buffers
#define NZ_OFF    (B_BASE + 2 * B_TILE_B)    // 47104
#define SMEM_B    (NZ_OFF + ROWS_WG * 4)     // 47360 B dynamic LDS

// ---------------------------------------------------------------------------
// Kernel 1: build Bfr = patches^T in TDM-friendly fragment-interleaved tiles.
// ---------------------------------------------------------------------------
__global__ void build_Bfr_kernel(const float* __restrict__ x,
                                 float* __restrict__ Bfr) {
    int idx = blockIdx.x * blockDim.x + threadIdx.x;   // < 4*144*3584
    int col = idx % 224;
    int rp  = (idx / 224) % 16;
    int T   = (idx / (224 * 16)) % NTILES;
    int c   = idx / (224 * 16 * NTILES);
    float v = 0.0f;
    if (col < 192) {
        int n = col >> 1;
        if (n < NPIX) {
            int p  = col & 1;
            int m  = T * BK + rp * 2 + p;
            int hp = (m / NWB) * PSZ + n / PSZ;
            int wp = (m % NWB) * PSZ + n % PSZ;
            v = x[((size_t)c * HP_DIM + hp) * WP_DIM + wp];
        }
    }
    Bfr[idx] = v;
}

// ---------------------------------------------------------------------------
// TDM issue helper: D# group0 built per call, group1 passed in (constant).
// CPOL is a template parameter: the builtin requires a constant immediate.
// ---------------------------------------------------------------------------
template <int CPOL>
__device__ __forceinline__ void tdm_load(unsigned lds_off,
                                         unsigned long long gaddr,
                                         v8i g1) {
    v4u g0;
    g0.x = 1u;                                        // count=1, user mode
    g0.y = lds_off;                                   // lds_addr
    g0.z = (unsigned)gaddr;                           // global_addr[31:0]
    g0.w = (unsigned)(gaddr >> 32) | (2u << 30);      // addr[56:32] | type=2
    v4i z4 = (v4i){0, 0, 0, 0};
#if __clang_major__ >= 23
    v8i z8 = (v8i){0, 0, 0, 0, 0, 0, 0, 0};
    __builtin_amdgcn_tensor_load_to_lds(g0, g1, z4, z4, z8, CPOL);
#else
    __builtin_amdgcn_tensor_load_to_lds(g0, g1, z4, z4, CPOL);
#endif
}

// ---------------------------------------------------------------------------
// Kernel 2: per (head, 64-row block): D(64x96) = A(64x4608) x B(4608x96).
// ---------------------------------------------------------------------------
__global__ __launch_bounds__(64)
void cross_head_gemm_kernel(const float* __restrict__ attn,
                            const float* __restrict__ Bfr,
                            const float* __restrict__ x,
                            float* __restrict__ out) {
    extern __shared__ char smem[];                    // offset 0 in LDS
    int* nz_lds = (int*)(smem + NZ_OFF);

    const int c     = blockIdx.y;
    const int row0  = blockIdx.x * ROWS_WG;
    const int tid   = threadIdx.x;                    // 0..63
    const int wave  = tid >> 5;
    const int lane  = tid & 31;
    const int laneh = lane >> 4;
    const int lanel = lane & 15;

    nz_lds[tid] = 0;

    // D# group1 for A: data_size=4B, pad 4 DWORDs per 32; tile 32x64 in a
    // 4608x4608 tensor (stride 4608).
    const v8i g1A = {(int)((2u << 16) | (1u << 20) | (4u << 22) | (3u << 25)),
                     (int)(0x1200u << 16), (int)(0x1200u << 16),
                     (int)(32u << 16), 64, 4608, 0, 0};
    // D# group1 for B: contiguous 3584-dword tile copy.
    const v8i g1B = {(int)(2u << 16), (int)0xE0000000u,
                     (int)(0x7u | (1u << 16)), (int)(3584u << 16),
                     1, 516096, 0, 0};

    const unsigned long long attnB =
        (unsigned long long)(size_t)attn +
        (((size_t)c * L_DIM + row0) * L_DIM) * 4ull;
    const unsigned long long bfrB =
        (unsigned long long)(size_t)Bfr + ((size_t)c * NTILES * B_TILE_F) * 4ull;

    const bool leader = (__builtin_amdgcn_readfirstlane((int)tid) < 32);

    v8f acc[2][6];
#pragma unroll
    for (int s = 0; s < 2; ++s)
#pragma unroll
        for (int nt = 0; nt < 6; ++nt)
            acc[s][nt] = (v8f){0.f, 0.f, 0.f, 0.f, 0.f, 0.f, 0.f, 0.f};

    int cnt0 = 0, cnt1 = 0;

    // fragment ping-pong slots (software pipeline across K-steps)
    v2f a0f[2], a1f[2], bf[2][6];
    const int aoff0 = (wave * 32 + lanel) * A_STR + 2 * laneh;
    const int aoff1 = aoff0 + 16 * A_STR;
    const int boff0 = (laneh * B_RP_STR + lanel) * 2;

    // prologue: DMA tile 0 into buffer 0 (A stream = non-temporal)
    if (leader) {
        tdm_load<1>(0u, attnB, g1A);
        tdm_load<0>((unsigned)B_BASE, bfrB, g1B);
    }
    __builtin_amdgcn_s_wait_tensorcnt(0);
    __syncthreads();

    for (int T = 0; T < NTILES; ++T) {
        const int par = T & 1;
        if (T + 1 < NTILES && leader) {
            const int np = (T + 1) & 1;
            tdm_load<1>((unsigned)(np * A_TILE_B),
                        attnB + (unsigned long long)(T + 1) * (BK * 4), g1A);
            tdm_load<0>((unsigned)(B_BASE + np * B_TILE_B),
                        bfrB + (unsigned long long)(T + 1) * B_TILE_B, g1B);
        }

        const float* Alds = (const float*)(smem + par * A_TILE_B);
        const float* Blds = (const float*)(smem + B_BASE + par * B_TILE_B);

        // preload K-step 0 fragments into slot 0
        a0f[0] = *(const v2f*)(Alds + aoff0);
        a1f[0] = *(const v2f*)(Alds + aoff1);
#pragma unroll
        for (int nt = 0; nt < 6; ++nt)
            bf[0][nt] = *(const v2f*)(Blds + boff0 + 32 * nt);

#pragma unroll
        for (int j = 0; j < 8; ++j) {
            const int cur = j & 1;
            const int nxt = cur ^ 1;
            if (j < 7) {   // preload next K-step while current computes
                const int ak = 4 * (j + 1);
                a0f[nxt] = *(const v2f*)(Alds + aoff0 + ak);
                a1f[nxt] = *(const v2f*)(Alds + aoff1 + ak);
#pragma unroll
                for (int nt = 0; nt < 6; ++nt)
                    bf[nxt][nt] = *(const v2f*)(Blds + boff0 + 32 * nt +
                                                (j + 1) * (2 * B_RP_STR * 2));
            }
            // fused count_nonzero (each attn element in exactly one fragment)
            cnt0 += (a0f[cur][0] != 0.0f) + (a0f[cur][1] != 0.0f);
            cnt1 += (a1f[cur][0] != 0.0f) + (a1f[cur][1] != 0.0f);
#pragma unroll
            for (int nt = 0; nt < 6; ++nt) {
                acc[0][nt] = __builtin_amdgcn_wmma_f32_16x16x4_f32(
                    false, a0f[cur], false, bf[cur][nt], (short)0,
                    acc[0][nt], false, false);
                acc[1][nt] = __builtin_amdgcn_wmma_f32_16x16x4_f32(
                    false, a1f[cur], false, bf[cur][nt], (short)0,
                    acc[1][nt], false, false);
            }
        }
        __builtin_amdgcn_s_wait_tensorcnt(0);
        __syncthreads();
    }

    // merge per-lane nonzero counts
    atomicAdd(&nz_lds[wave * 32 + lanel], cnt0);
    atomicAdd(&nz_lds[wave * 32 + 16 + lanel], cnt1);
    __syncthreads();

    // epilogue: normalize, fold (col2im), leaky_relu((att+1)*x)
#pragma unroll
    for (int s = 0; s < 2; ++s) {
#pragma unroll
        for (int nt = 0; nt < 6; ++nt) {
            const int k = 16 * nt + lanel;
            if (k < NPIX) {
#pragma unroll
                for (int r = 0; r < 8; ++r) {
                    const int lrow = wave * 32 + 16 * s + 8 * laneh + r;
                    const int l = row0 + lrow;
                    float nzf  = (float)nz_lds[lrow] + 1e-5f;
                    float attv = acc[s][nt][r] / nzf;
                    const int hp = (l / NWB) * PSZ + k / PSZ;
                    const int wp = (l % NWB) * PSZ + k % PSZ;
                    float xv  = x[((size_t)c * HP_DIM + hp) * WP_DIM + wp];
                    float val = (attv + 1.0f) * xv;
                    val = val >= 0.0f ? val : 0.2f * val;
                    out[((size_t)c * HP_DIM + hp) * WP_DIM + wp] = val;
                }
            }
        }
    }
}

// ---------------------------------------------------------------------------
extern "C" void kernel_launch(void* const* d_in, const int* in_sizes, int n_in,
                              void* d_out, int out_size, void* d_ws, size_t ws_size,
                              hipStream_t stream) {
    const float* x    = (const float*)d_in[0];   // [1,4,72,72,72]
    const float* attn = (const float*)d_in[1];   // [1,4,4608,4608]
    float* out = (float*)d_out;                  // [1,4,72,5184] flat
    float* Bfr = (float*)d_ws;                   // 4*144*3584 floats = 7.9 MB

    const int btotal = C_HEADS * NTILES * B_TILE_F;   // 2,064,384
    build_Bfr_kernel<<<btotal / 256, 256, 0, stream>>>(x, Bfr);

    dim3 grid(L_DIM / ROWS_WG, C_HEADS);         // 72 x 4 = 288 WGs
    cross_head_gemm_kernel<<<grid, 64, SMEM_B, stream>>>(attn, Bfr, x, out);
}